// PrototypeClassifier_14328010899989
// MI455X (gfx1250) — compile-verified
//
#include <hip/hip_runtime.h>
#include <hip/hip_bf16.h>
#include <math.h>

typedef float v2f __attribute__((ext_vector_type(2)));
typedef float v4f __attribute__((ext_vector_type(4)));
typedef float v8f __attribute__((ext_vector_type(8)));

#define D_LAT   40
#define N_PROTO 512
#define O_OUT   256

// ws float layout: [0..39] proto column sums, [40] ||proto||_F^2, [64..319] W row sums
__global__ void proto_precomp_kernel(const float* __restrict__ proto,
                                     const float* __restrict__ W,
                                     float* __restrict__ ws) {
    const int t = threadIdx.x;          // 256 threads, 1 block
    __shared__ float red[256];

    // per-dimension prototype sum (40 values)
    if (t < D_LAT) {
        float s = 0.f;
        #pragma unroll 8
        for (int p = 0; p < N_PROTO; ++p) s += proto[p * D_LAT + t];
        ws[t] = s;
    }

    // Frobenius sum of squares of prototypes
    float sq = 0.f;
    for (int i = t; i < N_PROTO * D_LAT; i += 256) { float v = proto[i]; sq += v * v; }
    red[t] = sq;
    __syncthreads();
    for (int off = 128; off > 0; off >>= 1) {
        if (t < off) red[t] += red[t + off];
        __syncthreads();
    }
    if (t == 0) ws[D_LAT] = red[0];

    // row sums of W (256 values)
    float s = 0.f;
    #pragma unroll 8
    for (int p = 0; p < N_PROTO; ++p) s += W[t * N_PROTO + p];
    ws[64 + t] = s;
}

// One wave handles 16 rows. dist for the 16 rows comes out of the diagonal of a
// V_WMMA_F32_16X16X4_F32 accumulation; the rest is pure NT streaming stores.
__global__ __launch_bounds__(256) void proto_main_kernel(
    const float* __restrict__ x, const float* __restrict__ bvec,
    const float* __restrict__ ws, float* __restrict__ xmat,
    float* __restrict__ out) {

    const int lane  = threadIdx.x & 31;
    const int wave  = threadIdx.x >> 5;
    const int gwave = blockIdx.x * 8 + wave;
    const size_t r0 = (size_t)gwave * 16;

    // ISA A-layout (16x4 f32): lane L holds row M = L&15, k = kstep*4 + (L>>4)*2 + {0,1}
    // ISA B-layout (4x16 f32): lane L holds col N = L&15, same k mapping.
    // We want B column n == y-row n, so the B fragment is a per-lane transform of A.
    const int m16 = lane & 15;
    const int kh  = (lane >> 4) * 2;

    const float psq = ws[D_LAT];

    v8f acc = {};
    const float* xrow = x + (r0 + (size_t)m16) * D_LAT;
    #pragma unroll
    for (int kk = 0; kk < 10; ++kk) {
        const int k = kk * 4 + kh;
        const float a0 = xrow[k], a1 = xrow[k + 1];
        const float s0 = ws[k],   s1 = ws[k + 1];
        v2f A  = {a0, a1};
        v2f Bf = {512.f * a0 - 2.f * s0, 512.f * a1 - 2.f * s1};
        // D[m,n] += sum_k x[m,k] * (512*x[n,k] - 2*psum[k]); diagonal is what we need
        acc = __builtin_amdgcn_wmma_f32_16x16x4_f32(
            false, A, false, Bf, (short)0, acc, false, false);
    }

    // Diagonal extraction per 16x16 f32 C/D layout:
    // row m: VGPR (m&7), lane m (m<8) or lane m+16 (m>=8)
    float dist[16];
    #pragma unroll
    for (int m = 0; m < 16; ++m) {
        const int src = (m < 8) ? m : (m + 16);
        const float d2 = __shfl(acc[m & 7], src, 32);
        dist[m] = sqrtf(d2 + psq);
    }

    // preload W row sums + bias fragments (2 float4 per lane covers 256 outputs)
    v4f w4[2], b4[2];
    #pragma unroll
    for (int j = 0; j < 2; ++j) {
        w4[j] = ((const v4f*)(ws + 64))[j * 32 + lane];
        b4[j] = ((const v4f*)bvec)[j * 32 + lane];
    }

    #pragma unroll
    for (int m = 0; m < 16; ++m) {
        const float d = dist[m];

        // xmat row: 512 floats = 128 v4f, 4 per lane, 512B contiguous per store
        v4f* xp = (v4f*)(xmat + (r0 + (size_t)m) * (size_t)N_PROTO);
        v4f dv = {d, d, d, d};
        #pragma unroll
        for (int j = 0; j < 4; ++j)
            __builtin_nontemporal_store(dv, xp + j * 32 + lane);

        // out row: 256 floats = 64 v4f, 2 per lane
        v4f* op = (v4f*)(out + (r0 + (size_t)m) * (size_t)O_OUT);
        #pragma unroll
        for (int j = 0; j < 2; ++j) {
            v4f r = d * w4[j] + b4[j];
            __builtin_nontemporal_store(r, op + j * 32 + lane);
        }
    }
}

extern "C" void kernel_launch(void* const* d_in, const int* in_sizes, int n_in,
                              void* d_out, int out_size, void* d_ws, size_t ws_size,
                              hipStream_t stream) {
    const float* x     = (const float*)d_in[0];
    const float* proto = (const float*)d_in[1];
    const float* W     = (const float*)d_in[2];
    const float* b     = (const float*)d_in[3];
    float* ws   = (float*)d_ws;

    const int B = in_sizes[0] / D_LAT;          // 262144
    float* xmat = (float*)d_out;                // [B, 512]
    float* out  = xmat + (size_t)B * N_PROTO;   // [B, 256]

    proto_precomp_kernel<<<1, 256, 0, stream>>>(proto, W, ws);

    // 8 waves/block * 16 rows/wave = 128 rows per block
    proto_main_kernel<<<B / 128, 256, 0, stream>>>(x, b, ws, xmat, out);
}